// CrossViewAttention_36498632081481
// MI455X (gfx1250) — compile-verified
//
#include <hip/hip_runtime.h>
#include <hip/hip_bf16.h>

typedef __attribute__((ext_vector_type(2))) float v2f;
typedef __attribute__((ext_vector_type(8))) float v8f;

#define N_Q     50000
#define N_KV    50000
#define N_E     500000
#define DIM     128
#define ALPHA_L 0.2f
#define EPS_F   1e-10f

// workspace layout (float offsets)
#define OFF_WAQ  0                       // 128
#define OFF_WAK  128                     // 128
#define OFF_CST  256                     // 2: c_q, c_k
#define OFF_SQ   512                     // N_Q
#define OFF_SK   (OFF_SQ + N_Q)          // N_KV
#define OFF_EMAX (OFF_SK + N_KV)         // N_Q
#define OFF_ESUM (OFF_EMAX + N_Q)        // N_Q
#define OFF_EBUF (OFF_ESUM + N_Q)        // N_E
#define OFF_KVP  (OFF_EBUF + N_E)        // N_KV*DIM  (multiple of 4 -> float4 ok)

// ---------------- init: zero accumulators, -inf for segment max ----------------
__global__ void k_init(float* __restrict__ ws, float* __restrict__ out) {
  int i = blockIdx.x * blockDim.x + threadIdx.x;
  if (i < N_Q * DIM) out[i] = 0.0f;
  if (i < N_Q) {
    ws[OFF_EMAX + i] = -INFINITY;   // bit pattern 0xFF800000, works for int-max/uint-min trick
    ws[OFF_ESUM + i] = 0.0f;
  }
}

// ---------------- fold attention vector through projection ----------------
// w_aq[i] = sum_o W[o,i]*a[o] ; w_ak[i] = sum_o W[o,i]*a[128+o]
// c_q = b . a[:128] ; c_k = b . a[128:]
__global__ __launch_bounds__(256) void k_att(const float* __restrict__ W,
                                             const float* __restrict__ b,
                                             const float* __restrict__ a,
                                             float* __restrict__ ws) {
  int t = threadIdx.x;
  if (t < 128) {
    float s = 0.f;
    for (int o = 0; o < DIM; ++o) s += W[o * DIM + t] * a[o];
    ws[OFF_WAQ + t] = s;
  } else {
    int i = t - 128;
    float s = 0.f;
    for (int o = 0; o < DIM; ++o) s += W[o * DIM + i] * a[DIM + o];
    ws[OFF_WAK + i] = s;
  }
  if (t == 0) { float s = 0.f; for (int o = 0; o < DIM; ++o) s += b[o] * a[o];       ws[OFF_CST + 0] = s; }
  if (t == 1) { float s = 0.f; for (int o = 0; o < DIM; ++o) s += b[o] * a[DIM + o]; ws[OFF_CST + 1] = s; }
}

// ---------------- per-node score scalar: s[n] = X[n] . w + c  (wave per node) ----------------
__global__ __launch_bounds__(256) void k_scal(const float* __restrict__ X,
                                              const float* __restrict__ wvec,
                                              const float* __restrict__ cptr,
                                              float* __restrict__ sout, int n) {
  int gw   = (blockIdx.x * blockDim.x + threadIdx.x) >> 5;
  int lane = threadIdx.x & 31;
  if (gw >= n) return;
  float4 x = ((const float4*)(X + (size_t)gw * DIM))[lane];
  float4 w = ((const float4*)wvec)[lane];
  float s = x.x * w.x + x.y * w.y + x.z * w.z + x.w * w.w;
  #pragma unroll
  for (int m = 16; m >= 1; m >>= 1) s += __shfl_xor(s, m, 32);
  if (lane == 0) sout[gw] = s + cptr[0];
}

// ---------------- WMMA fp32 projection: P = X @ W^T + b  (wave per 16-row tile) ----------------
__global__ __launch_bounds__(256) void k_proj(const float* __restrict__ X,
                                              const float* __restrict__ W,
                                              const float* __restrict__ bias,
                                              float* __restrict__ P, int nrows) {
  int wave = threadIdx.x >> 5;
  int row0 = (blockIdx.x * 8 + wave) * 16;
  if (row0 + 16 > nrows) return;
  int lane = threadIdx.x & 31;
  int half = lane >> 4;          // 0: K+0/1 , 1: K+2/3  (A frag layout, ISA 7.12.2)
  int l15  = lane & 15;
  int koff = half * 2;
  const float* Xr = X + (size_t)(row0 + l15) * DIM;   // A: row m = l15
  #pragma unroll 1
  for (int ct = 0; ct < 8; ++ct) {
    int col0 = ct * 16;
    const float* Wr = W + (size_t)(col0 + l15) * DIM; // B[k][n] = W[n][k]
    v8f acc = {};
    #pragma unroll
    for (int k0 = 0; k0 < DIM; k0 += 4) {
      v2f a = *(const v2f*)(Xr + k0 + koff);
      v2f b = *(const v2f*)(Wr + k0 + koff);
      acc = __builtin_amdgcn_wmma_f32_16x16x4_f32(false, a, false, b,
                                                  (short)0, acc, false, false);
    }
    float bv = bias[col0 + l15];
    float* Pout = P + col0 + l15;
    #pragma unroll
    for (int r = 0; r < 8; ++r) {                      // D elem (m,n): vgpr r, lane n+16*(m/8)
      int m = r + half * 8;
      Pout[(size_t)(row0 + m) * DIM] = acc[r] + bv;
    }
  }
}

// ---------------- edge scores + segment max (bit-punned float atomic max) ----------------
__global__ __launch_bounds__(256) void k_edge1(const long long* __restrict__ eidx,
                                               const float* __restrict__ ws_ro,
                                               const float* __restrict__ attb,
                                               float* __restrict__ ws) {
  int e = blockIdx.x * blockDim.x + threadIdx.x;
  if (e >= N_E) return;
  long long q = eidx[e];
  long long k = eidx[N_E + e];
  float v = ws_ro[OFF_SQ + q] + ws_ro[OFF_SK + k] + attb[0];
  v = v > 0.f ? v : ALPHA_L * v;        // leaky relu
  ws[OFF_EBUF + e] = v;
  float* addr = ws + OFF_EMAX + q;
  if (v >= 0.f) atomicMax((int*)addr, __float_as_int(v));
  else          atomicMin((unsigned int*)addr, __float_as_uint(v));
}

// ---------------- exp + segment sum + fused weighted aggregate (wave per edge) ----------------
__global__ __launch_bounds__(256) void k_edge2(const long long* __restrict__ eidx,
                                               float* __restrict__ ws,
                                               float* __restrict__ out) {
  int gw   = (blockIdx.x * blockDim.x + threadIdx.x) >> 5;
  int lane = threadIdx.x & 31;
  if (gw >= N_E) return;
  long long q = eidx[gw];
  long long k = eidx[N_E + gw];
  float w = __expf(ws[OFF_EBUF + gw] - ws[OFF_EMAX + q]);
  if (lane == 0) unsafeAtomicAdd(ws + OFF_ESUM + q, w);
  float4 kv = ((const float4*)(ws + OFF_KVP + (size_t)k * DIM))[lane];
  float* o = out + (size_t)q * DIM + lane * 4;
  unsafeAtomicAdd(o + 0, w * kv.x);
  unsafeAtomicAdd(o + 1, w * kv.y);
  unsafeAtomicAdd(o + 2, w * kv.z);
  unsafeAtomicAdd(o + 3, w * kv.w);
}

// ---------------- normalize by (segment sum + EPS) ----------------
__global__ __launch_bounds__(256) void k_norm(const float* __restrict__ ws,
                                              float* __restrict__ out) {
  int i = blockIdx.x * blockDim.x + threadIdx.x;
  if (i >= N_Q * DIM) return;
  out[i] = out[i] / (ws[OFF_ESUM + (i >> 7)] + EPS_F);
}

extern "C" void kernel_launch(void* const* d_in, const int* in_sizes, int n_in,
                              void* d_out, int out_size, void* d_ws, size_t ws_size,
                              hipStream_t stream) {
  const float*     qn   = (const float*)d_in[0];      // [N_Q, D]
  const float*     kvn  = (const float*)d_in[1];      // [N_KV, D]
  const long long* eidx = (const long long*)d_in[2];  // [2, E] int64
  const float*     pw   = (const float*)d_in[3];      // [D, D]
  const float*     pb   = (const float*)d_in[4];      // [D]
  const float*     aw   = (const float*)d_in[5];      // [1, 2D]
  const float*     ab   = (const float*)d_in[6];      // [1]
  float* out = (float*)d_out;
  float* ws  = (float*)d_ws;

  // 1. init accumulators / output
  k_init<<<(N_Q * DIM + 255) / 256, 256, 0, stream>>>(ws, out);
  // 2. fold attention vector through projection
  k_att<<<1, 256, 0, stream>>>(pw, pb, aw, ws);
  // 3. per-node score scalars (wave per node)
  k_scal<<<(N_Q * 32 + 255) / 256, 256, 0, stream>>>(qn,  ws + OFF_WAQ, ws + OFF_CST,     ws + OFF_SQ, N_Q);
  k_scal<<<(N_KV * 32 + 255) / 256, 256, 0, stream>>>(kvn, ws + OFF_WAK, ws + OFF_CST + 1, ws + OFF_SK, N_KV);
  // 4. kv projection via fp32 WMMA (needed for aggregation)
  k_proj<<<(N_KV + 127) / 128, 256, 0, stream>>>(kvn, pw, pb, ws + OFF_KVP, N_KV);
  // 5. edge scores + segment max
  k_edge1<<<(N_E + 255) / 256, 256, 0, stream>>>(eidx, ws, ab, ws);
  // 6. exp + segment sum + fused weighted aggregation (wave per edge)
  k_edge2<<<(N_E * 32 + 255) / 256 > 65535 * 256 ? (N_E * 32 + 255) / 256 : (N_E * 32 + 255) / 256,
            256, 0, stream>>>(eidx, ws, out);
  // 7. normalize
  k_norm<<<(N_Q * DIM + 255) / 256, 256, 0, stream>>>(ws, out);
}